// DMPNN_30777735643630
// MI455X (gfx1250) — compile-verified
//
#include <hip/hip_runtime.h>

// ---------------------------------------------------------------------------
// DMPNN on MI455X (gfx1250).
// N=4096, HIDDEN=16, EDGE=3, LAYERS=3. adj is a dense 0/1 mask, ~0.78% dense.
// Strategy:
//   * Pass 1 exploits sparsity: stream adj (67MB), gather edge rows only at
//     nonzeros (~17MB of lines) -> deg_in + esum.  Deterministic 2-pass reduce.
//   * Per layer: adjT @ (h W_h^T) done with V_WMMA_F32_16X16X4_F32 (exact
//     fp32), B-operand chunk staged in LDS, K split 8 ways, partials reduced
//     in a fused update kernel that also applies esum*We^T + deg*Wb and U.
//   * adj (67MB) stays resident in the 192MB L2 across the 3 layers.
// ---------------------------------------------------------------------------

#define GN      4096
#define GH      16
#define AGG_CHUNKS   32   // rows per chunk = 128
#define AGG_ROWS     128
#define K_CHUNKS     8    // K rows per chunk = 512
#define K_ROWS       512

typedef __attribute__((ext_vector_type(2))) float v2f;
typedef __attribute__((ext_vector_type(8))) float v8f;

// ---- Pass 1a: per-chunk deg/esum partials (sparse gather of edge) ----------
__global__ __launch_bounds__(256) void k_agg_part(const float* __restrict__ adj,
                                                  const float* __restrict__ edge,
                                                  float4* __restrict__ part) {
  const int col   = blockIdx.x * 256 + threadIdx.x;   // 0..4095 (coalesced)
  const int chunk = blockIdx.y;                        // 0..31
  const int r0 = chunk * AGG_ROWS;
  float d = 0.f, e0 = 0.f, e1 = 0.f, e2 = 0.f;
  for (int r = r0; r < r0 + AGG_ROWS; ++r) {
    float a = adj[(size_t)r * GN + col];
    if (a != 0.0f) {                                   // ~0.78% taken: skip edge gather
      const float* ep = edge + ((size_t)r * GN + col) * 3;
      d  += a;
      e0 += a * ep[0];
      e1 += a * ep[1];
      e2 += a * ep[2];
    }
  }
  part[(size_t)chunk * GN + col] = make_float4(d, e0, e1, e2);
}

// ---- Pass 1b: deterministic reduce over chunks -----------------------------
__global__ __launch_bounds__(256) void k_agg_reduce(const float4* __restrict__ part,
                                                    float4* __restrict__ aggr) {
  const int col = blockIdx.x * 256 + threadIdx.x;
  float4 s = make_float4(0.f, 0.f, 0.f, 0.f);
  for (int c = 0; c < AGG_CHUNKS; ++c) {
    float4 p = part[(size_t)c * GN + col];
    s.x += p.x; s.y += p.y; s.z += p.z; s.w += p.w;
  }
  aggr[col] = s;
}

// ---- Node embedding: h = nf * emb_w^T + emb_b ------------------------------
__global__ __launch_bounds__(256) void k_emb(const float* __restrict__ nf,
                                             const float* __restrict__ w,
                                             const float* __restrict__ b,
                                             float* __restrict__ h) {
  const int idx = blockIdx.x * 256 + threadIdx.x;      // 0..65535
  const int i = idx >> 4, k = idx & 15;
  h[idx] = nf[i] * w[k] + b[k];
}

// ---- M = h @ W_h^T  (4096x16x16) -------------------------------------------
__global__ __launch_bounds__(256) void k_matM(const float* __restrict__ h,
                                              const float* __restrict__ Wh,   // [16][16]
                                              float* __restrict__ M) {
  __shared__ float sW[256];
  sW[threadIdx.x] = Wh[threadIdx.x];
  __syncthreads();
  const int idx = blockIdx.x * 256 + threadIdx.x;
  const int i = idx >> 4, k = idx & 15;
  const float* hr = h + i * GH;
  float s = 0.f;
#pragma unroll
  for (int d = 0; d < GH; ++d) s += hr[d] * sW[k * GH + d];
  M[idx] = s;
}

// ---- SpMM via WMMA: part[c][j][k] = sum_{i in chunk c} adj[i][j] * M[i][k] -
// Block = 8 waves = 8 adjacent 16-row j-tiles, all on the same K-chunk.
// B operand (M chunk, 512x16 = 32KB) staged in LDS, shared by all 8 waves.
__global__ __launch_bounds__(256) void k_spmm(const float* __restrict__ adj,
                                              const float* __restrict__ M,
                                              float* __restrict__ part) {
  __shared__ float sM[K_ROWS * GH];                    // 32 KB
  const int chunk = blockIdx.x >> 5;                   // 0..7
  const int tgrp  = blockIdx.x & 31;                   // 0..31
  const int tid   = threadIdx.x;

  // cooperative LDS fill: 8192 floats = 2048 float4
  const float4* Mv  = (const float4*)(M + (size_t)chunk * K_ROWS * GH);
  float4*       sMv = (float4*)sM;
  for (int i = tid; i < (K_ROWS * GH) / 4; i += 256) sMv[i] = Mv[i];
  __syncthreads();

  const int wave = tid >> 5;
  const int lane = tid & 31;
  const int tile = tgrp * 8 + wave;                    // 0..255
  const int j0   = tile * 16;
  const int n    = lane & 15;
  const int koff = (lane < 16) ? 0 : 2;                // A/B K-split per ISA layout
  const int kbase = chunk * K_ROWS;

  v8f acc = {};
  const float* a0 = adj + (size_t)(kbase + koff) * GN + j0 + n;  // A: adjT tile
  const float* b0 = sM + koff * GH + n;                          // B: M rows in LDS

#pragma unroll 4
  for (int kk = 0; kk < K_ROWS; kk += 4) {
    v2f a, b;
    a.x = a0[(size_t)kk * GN];
    a.y = a0[(size_t)kk * GN + GN];
    b.x = b0[kk * GH];
    b.y = b0[kk * GH + GH];
    acc = __builtin_amdgcn_wmma_f32_16x16x4_f32(
        /*neg_a=*/false, a, /*neg_b=*/false, b,
        /*c_mod=*/(short)0, acc, /*reuse_a=*/false, /*reuse_b=*/false);
  }

  // C/D layout: VGPR v -> row (v + 8*(lane>=16)), col = lane&15
  float* p = part + ((size_t)chunk * GN + j0) * GH + n;
  const int rbase = (lane < 16) ? 0 : 8;
#pragma unroll
  for (int v = 0; v < 8; ++v) p[(size_t)(rbase + v) * GH] = acc[v];
}

// ---- Fused: reduce partials, add edge/deg terms, apply U, write h' ---------
__global__ __launch_bounds__(256) void k_update(const float* __restrict__ h,
                                                const float* __restrict__ part,  // [8][4096][16]
                                                const float4* __restrict__ aggr, // (deg,e0,e1,e2)
                                                const float* __restrict__ We,    // [16][3]
                                                const float* __restrict__ Wb,    // [16]
                                                const float* __restrict__ Uw,    // [16][16]
                                                const float* __restrict__ Ub,    // [16]
                                                float* __restrict__ hout) {
  const int j = blockIdx.x * 256 + threadIdx.x;        // 0..4095
  const float4 ag = aggr[j];
  float t[GH];
#pragma unroll
  for (int k = 0; k < GH; ++k) {
    float m = ag.x * Wb[k] + ag.y * We[k * 3 + 0] + ag.z * We[k * 3 + 1] + ag.w * We[k * 3 + 2];
    t[k] = h[(size_t)j * GH + k] + m;
  }
#pragma unroll
  for (int c = 0; c < K_CHUNKS; ++c) {
    const float* pp = part + ((size_t)c * GN + j) * GH;
#pragma unroll
    for (int k = 0; k < GH; ++k) t[k] += pp[k];
  }
#pragma unroll
  for (int k2 = 0; k2 < GH; ++k2) {
    float s = Ub[k2];
#pragma unroll
    for (int k = 0; k < GH; ++k) s += t[k] * Uw[k2 * GH + k];
    hout[(size_t)j * GH + k2] = s;
  }
}

// ---- Readout: sum_j h[j] . ro_w + ro_b  (single block, deterministic) ------
__global__ __launch_bounds__(256) void k_readout(const float* __restrict__ h,
                                                 const float* __restrict__ row,
                                                 const float* __restrict__ rob,
                                                 float* __restrict__ out) {
  __shared__ float red[256];
  const int tid = threadIdx.x;
  float rw[GH];
#pragma unroll
  for (int k = 0; k < GH; ++k) rw[k] = row[k];
  float p = 0.f;
  for (int j = tid; j < GN; j += 256) {
#pragma unroll
    for (int k = 0; k < GH; ++k) p += h[(size_t)j * GH + k] * rw[k];
  }
  red[tid] = p;
  __syncthreads();
  for (int s = 128; s > 0; s >>= 1) {
    if (tid < s) red[tid] += red[tid + s];
    __syncthreads();
  }
  if (tid == 0) out[0] = red[0] + rob[0];
}

// ---------------------------------------------------------------------------
extern "C" void kernel_launch(void* const* d_in, const int* in_sizes, int n_in,
                              void* d_out, int out_size, void* d_ws, size_t ws_size,
                              hipStream_t stream) {
  (void)in_sizes; (void)n_in; (void)out_size; (void)ws_size;
  const float* nf    = (const float*)d_in[0];
  const float* edge  = (const float*)d_in[1];
  const float* adj   = (const float*)d_in[2];
  const float* emb_w = (const float*)d_in[3];
  const float* emb_b = (const float*)d_in[4];
  const float* W_h   = (const float*)d_in[5];   // [3][16][16]
  const float* W_e   = (const float*)d_in[6];   // [3][16][3]
  const float* W_b   = (const float*)d_in[7];   // [3][16]
  const float* U_w   = (const float*)d_in[8];   // [3][16][16]
  const float* U_b   = (const float*)d_in[9];   // [3][16]
  const float* ro_w  = (const float*)d_in[10];  // [16]
  const float* ro_b  = (const float*)d_in[11];  // [1]
  float* out = (float*)d_out;

  // Workspace carve-up (~2.9 MB total):
  //   [0, 2MB)        : agg partials (32*4096*float4), then reused for
  //                     spmm partials (8*4096*16 floats) — disjoint lifetimes.
  //   [2MB, +64KB)    : aggr (deg,e0,e1,e2) per column
  //   then h0, h1, M  : 256KB each
  char* ws = (char*)d_ws;
  float*  scratchA = (float*)ws;                              // 2 MB region
  float4* aggr     = (float4*)(ws + (2u << 20));
  float*  h0       = (float*)(ws + (2u << 20) + (64u << 10));
  float*  h1       = h0 + GN * GH;
  float*  M        = h1 + GN * GH;

  // Pass 1: deg_in + esum (sparse gather)
  k_agg_part  <<<dim3(GN / 256, AGG_CHUNKS), 256, 0, stream>>>(adj, edge, (float4*)scratchA);
  k_agg_reduce<<<GN / 256, 256, 0, stream>>>((const float4*)scratchA, aggr);

  // h = embed(node_features)
  k_emb<<<(GN * GH) / 256, 256, 0, stream>>>(nf, emb_w, emb_b, h0);

  float* hc = h0;
  float* hn = h1;
  for (int l = 0; l < 3; ++l) {
    k_matM  <<<(GN * GH) / 256, 256, 0, stream>>>(hc, W_h + l * 256, M);
    k_spmm  <<<K_CHUNKS * 32, 256, 0, stream>>>(adj, M, scratchA);
    k_update<<<GN / 256, 256, 0, stream>>>(hc, scratchA, aggr,
                                           W_e + l * 48, W_b + l * 16,
                                           U_w + l * 256, U_b + l * 16, hn);
    float* tmp = hc; hc = hn; hn = tmp;
  }

  k_readout<<<1, 256, 0, stream>>>(hc, ro_w, ro_b, out);
}